// Agent_214748364878
// MI455X (gfx1250) — compile-verified
//
#include <hip/hip_runtime.h>
#include <math.h>
#include <stdint.h>

// Problem constants (from reference)
#define B_  64
#define M_  5
#define N_  1000
#define D_  128
#define H_  8
#define KS_ 16
#define F_  8      // ndf feature width

typedef float v2f __attribute__((ext_vector_type(2)));
typedef float v8f __attribute__((ext_vector_type(8)));

__device__ __forceinline__ float wave_sum(float v) {
#pragma unroll
  for (int o = 16; o > 0; o >>= 1) v += __shfl_down(v, o, 32);
  return v;
}
__device__ __forceinline__ float wave_max(float v) {
#pragma unroll
  for (int o = 16; o > 0; o >>= 1) v = fmaxf(v, __shfl_down(v, o, 32));
  return v;
}
template <int NW>
__device__ __forceinline__ float block_sum(float v, float* scratch) {
  v = wave_sum(v);
  int w = threadIdx.x >> 5, l = threadIdx.x & 31;
  __syncthreads();
  if (l == 0) scratch[w] = v;
  __syncthreads();
  float r = 0.f;
#pragma unroll
  for (int i = 0; i < NW; ++i) r += scratch[i];
  return r;
}
template <int NW>
__device__ __forceinline__ float block_max(float v, float* scratch) {
  v = wave_max(v);
  int w = threadIdx.x >> 5, l = threadIdx.x & 31;
  __syncthreads();
  if (l == 0) scratch[w] = v;
  __syncthreads();
  float r = scratch[0];
#pragma unroll
  for (int i = 1; i < NW; ++i) r = fmaxf(r, scratch[i]);
  return r;
}

// ---------------------------------------------------------------------------
// K1: query[b,m,d] = fixed_context + W_pcv @ [prev_emb, veh_dyn]
//     qk_w[b,m,h,j] = sum_k query[b,m,h*16+k] * W_pns[128 + h*16+k][j]
// grid = B*M blocks of 128
__global__ void __launch_bounds__(128)
k1_query_qkw(const float* __restrict__ fixed_context,
             const float* __restrict__ prev,
             const float* __restrict__ veh,
             const float* __restrict__ W_pcv,
             const float* __restrict__ W_pns,
             float* __restrict__ ws_query,
             float* __restrict__ ws_qkw) {
  int bm = blockIdx.x;
  int b  = bm / M_;
  int d  = threadIdx.x;
  __shared__ float qs[D_];
  const float* pr = prev + (size_t)bm * D_;
  const float* wv = W_pcv + (size_t)d * (D_ + 3);
  float acc = fixed_context[(size_t)b * D_ + d];
#pragma unroll 4
  for (int j = 0; j < D_; ++j) acc += wv[j] * pr[j];
  const float* vd = veh + (size_t)bm * 3;
  acc += wv[D_ + 0] * vd[0] + wv[D_ + 1] * vd[1] + wv[D_ + 2] * vd[2];
  ws_query[(size_t)bm * D_ + d] = acc;
  qs[d] = acc;
  __syncthreads();
  if (d < H_ * F_) {
    int h = d >> 3, j = d & 7;
    float a2 = 0.f;
#pragma unroll
    for (int k = 0; k < KS_; ++k)
      a2 += qs[h * KS_ + k] * W_pns[(size_t)(D_ + h * KS_ + k) * F_ + j];
    ws_qkw[((size_t)bm * H_ + h) * F_ + j] = a2;
  }
}

// ---------------------------------------------------------------------------
// K2: one block per (h,b). Phase 1: compat = (Q.Kstat [WMMA f32 16x16x4 chain]
//     + ndf.qk_w)/4 + mask_log -> ws, track max. Phase 2: softmax-weighted
//     sums over V_static and ndf -> concat rows (with W_pns_V correction).
__global__ void __launch_bounds__(128)
k2_attention(const float* __restrict__ gK,   // glimpse_K_static [H,B,N,16]
             const float* __restrict__ gV,   // glimpse_V_static [H,B,N,16]
             const float* __restrict__ ndf,  // [B,M,N,8]
             const unsigned char* __restrict__ mask,  // [B,M,N] (numpy bool)
             const float* __restrict__ W_pns,
             const float* __restrict__ ws_query,
             const float* __restrict__ ws_qkw,
             float* __restrict__ ws_compat,  // [H,B,M,N]
             float* __restrict__ ws_concat)  // [B,M,D]
{
  int hb = blockIdx.x;
  int h = hb / B_, b = hb % B_;
  __shared__ float Qs[16 * 17];
  __shared__ float qkw[M_ * F_];
  __shared__ float red[4];
  __shared__ float rv[25];
  int tid = threadIdx.x;

  for (int i = tid; i < 16 * KS_; i += 128) {
    int m = i >> 4, k = i & 15;
    Qs[m * 17 + k] =
        (m < M_) ? ws_query[(size_t)(b * M_ + m) * D_ + h * KS_ + k] : 0.f;
  }
  if (tid < M_ * F_) {
    int m = tid >> 3, j = tid & 7;
    qkw[tid] = ws_qkw[((size_t)(b * M_ + m) * H_ + h) * F_ + j];
  }
  __syncthreads();

  int wave = tid >> 5, lane = tid & 31;
  int row = lane & 15, off = (lane < 16) ? 0 : 2;
  // A operand (16x4 f32 layout): lanes 0-15 -> K=k0,k0+1; lanes 16-31 -> K=k0+2,k0+3
  v2f a0, a1, a2, a3;
  a0.x = Qs[row * 17 + 0 + off];  a0.y = Qs[row * 17 + 1 + off];
  a1.x = Qs[row * 17 + 4 + off];  a1.y = Qs[row * 17 + 5 + off];
  a2.x = Qs[row * 17 + 8 + off];  a2.y = Qs[row * 17 + 9 + off];
  a3.x = Qs[row * 17 + 12 + off]; a3.y = Qs[row * 17 + 13 + off];

  const float* Kbase = gK + (size_t)hb * N_ * KS_;
  float* cmp = ws_compat + (size_t)hb * M_ * N_;
  float lmax = -__builtin_inff();
  const int NT = (N_ + 15) / 16;  // 63 tiles, wave-uniform loop (EXEC all-1s)
  for (int t = wave; t < NT; t += 4) {
    int n0 = t * 16;
    int nn = n0 + row;
    int nc = nn < N_ ? nn : (N_ - 1);
    const float* kr = Kbase + (size_t)nc * KS_;
    v2f b0, b1, b2, b3;
    b0.x = kr[0 + off];  b0.y = kr[1 + off];
    b1.x = kr[4 + off];  b1.y = kr[5 + off];
    b2.x = kr[8 + off];  b2.y = kr[9 + off];
    b3.x = kr[12 + off]; b3.y = kr[13 + off];
    v8f c = {};
    c = __builtin_amdgcn_wmma_f32_16x16x4_f32(false, a0, false, b0, (short)0, c, false, false);
    c = __builtin_amdgcn_wmma_f32_16x16x4_f32(false, a1, false, b1, (short)0, c, false, false);
    c = __builtin_amdgcn_wmma_f32_16x16x4_f32(false, a2, false, b2, (short)0, c, false, false);
    c = __builtin_amdgcn_wmma_f32_16x16x4_f32(false, a3, false, b3, (short)0, c, false, false);
    if (lane < 16 && nn < N_) {
#pragma unroll
      for (int m = 0; m < M_; ++m) {
        const float* fr = ndf + ((size_t)(b * M_ + m) * N_ + nn) * F_;
        float dot = 0.f;
#pragma unroll
        for (int j = 0; j < F_; ++j) dot += qkw[m * F_ + j] * fr[j];
        float val = (c[m] + dot) * 0.25f;  // 1/sqrt(KS)
        if (!mask[(size_t)(b * M_ + m) * N_ + nn]) val = -__builtin_inff();
        cmp[(size_t)m * N_ + nn] = val;
        lmax = fmaxf(lmax, val);
      }
    }
  }
  __threadfence();
  float gmax = block_max<4>(lmax, red);

  const float* Vbase = gV + (size_t)hb * N_ * KS_;
#pragma unroll 1
  for (int m = 0; m < M_; ++m) {
    float aS = 0.f;
    float aV[KS_], aF[F_];
#pragma unroll
    for (int k = 0; k < KS_; ++k) aV[k] = 0.f;
#pragma unroll
    for (int j = 0; j < F_; ++j) aF[j] = 0.f;
    const float* cm = cmp + (size_t)m * N_;
    const float* fm = ndf + (size_t)(b * M_ + m) * N_ * F_;
    for (int n = tid; n < N_; n += 128) {
      float e = __expf(cm[n] - gmax);  // masked (-inf) -> 0
      aS += e;
      const float* vr = Vbase + (size_t)n * KS_;
#pragma unroll
      for (int k = 0; k < KS_; ++k) aV[k] += e * vr[k];
      const float* fr = fm + (size_t)n * F_;
#pragma unroll
      for (int j = 0; j < F_; ++j) aF[j] += e * fr[j];
    }
#pragma unroll
    for (int i = 0; i < 25; ++i) {
      float v = (i == 0) ? aS : (i < 17 ? aV[i - 1] : aF[i - 17]);
      float r = block_sum<4>(v, red);
      if (tid == 0) rv[i] = r;
    }
    __syncthreads();
    if (tid < KS_) {
      float corr = 0.f;
#pragma unroll
      for (int j = 0; j < F_; ++j)
        corr += W_pns[(size_t)(h * KS_ + tid) * F_ + j] * rv[17 + j];
      ws_concat[(size_t)(b * M_ + m) * D_ + h * KS_ + tid] =
          (rv[1 + tid] + corr) / rv[0];
    }
    __syncthreads();
  }
}

// ---------------------------------------------------------------------------
// K3: final_Q = concat @ po^T ; ql_w[b,m,j] = sum_d final_Q[d]*W_pns[256+d][j]
__global__ void __launch_bounds__(128)
k3_finalq_qlw(const float* __restrict__ ws_concat,
              const float* __restrict__ po,
              const float* __restrict__ W_pns,
              float* __restrict__ ws_fq,
              float* __restrict__ ws_qlw) {
  int bm = blockIdx.x;
  int d = threadIdx.x;
  __shared__ float cs[D_];
  __shared__ float fs[D_];
  cs[d] = ws_concat[(size_t)bm * D_ + d];
  __syncthreads();
  const float* pr = po + (size_t)d * D_;
  float acc = 0.f;
#pragma unroll 4
  for (int j = 0; j < D_; ++j) acc += pr[j] * cs[j];
  ws_fq[(size_t)bm * D_ + d] = acc;
  fs[d] = acc;
  __syncthreads();
  if (d < F_) {
    float a2 = 0.f;
    for (int k = 0; k < D_; ++k)
      a2 += fs[k] * W_pns[(size_t)(2 * D_ + k) * F_ + d];
    ws_qlw[(size_t)bm * F_ + d] = a2;
  }
}

// ---------------------------------------------------------------------------
// K4: logits via WMMA f32 16x16x4 chained over K=128.
// logit_K tile is staged with async global->LDS DMA (ASYNCcnt path):
// one GLOBAL_LOAD_ASYNC_TO_LDS_B128 per 16B chunk, no VGPR round-trip.
// grid = (16 n-groups of 64, B). Block = 128 (4 waves, one 16-n tile each).
__global__ void __launch_bounds__(128)
k4_logits(const float* __restrict__ lK,   // [B,N,D]
          const float* __restrict__ ndf,
          const unsigned char* __restrict__ mask,
          const float* __restrict__ ws_fq,
          const float* __restrict__ ws_qlw,
          float* __restrict__ ws_logits) {
  int b = blockIdx.y;
  int n_base = blockIdx.x * 64;
  __shared__ float fqs[16 * 132];   // padded stride vs 64 banks
  __shared__ float lks[64 * 132];
  __shared__ float qlw[M_ * F_];
  int tid = threadIdx.x;

  for (int i = tid; i < 16 * D_; i += 128) {
    int m = i >> 7, kk = i & 127;
    fqs[m * 132 + kk] = (m < M_) ? ws_fq[(size_t)(b * M_ + m) * D_ + kk] : 0.f;
  }
  if (tid < M_ * F_) qlw[tid] = ws_qlw[(size_t)b * M_ * F_ + tid];

  // Stage logit_K tile [64 rows x 128 floats] into LDS via async DMA.
  const float* lKb = lK + (size_t)b * N_ * D_;
  for (int i = tid; i < 64 * (D_ / 4); i += 128) {  // 16-byte chunks
    int r = i >> 5, cq = (i & 31) * 4;
    int n = n_base + r;
    float* ldst = &lks[r * 132 + cq];
    if (n < N_) {
      // low 32 bits of a flat LDS pointer == LDS byte offset (aperture rule)
      uint32_t lds_off = (uint32_t)(uintptr_t)ldst;
      const float* gsrc = lKb + (size_t)n * D_ + cq;
      asm volatile("global_load_async_to_lds_b128 %0, %1, off"
                   :: "v"(lds_off), "v"(gsrc)
                   : "memory");
    } else {
      ldst[0] = 0.f; ldst[1] = 0.f; ldst[2] = 0.f; ldst[3] = 0.f;
    }
  }
  asm volatile("s_wait_asynccnt 0x0" ::: "memory");
  __syncthreads();

  int wave = tid >> 5, lane = tid & 31;
  int row = lane & 15, off = (lane < 16) ? 0 : 2;
  int rl = wave * 16 + row;
  const float* arow = fqs + row * 132;
  const float* brow = lks + rl * 132;
  v8f c = {};
#pragma unroll
  for (int s = 0; s < 32; ++s) {
    int k0 = s * 4 + off;
    v2f av; av.x = arow[k0]; av.y = arow[k0 + 1];
    v2f bv; bv.x = brow[k0]; bv.y = brow[k0 + 1];
    c = __builtin_amdgcn_wmma_f32_16x16x4_f32(false, av, false, bv, (short)0, c, false, false);
  }
  int nn = n_base + wave * 16 + row;
  if (lane < 16 && nn < N_) {
    const float inv = 0.08838834764831845f;  // 1/sqrt(128)
#pragma unroll
    for (int m = 0; m < M_; ++m) {
      const float* fr = ndf + ((size_t)(b * M_ + m) * N_ + nn) * F_;
      float dot = 0.f;
#pragma unroll
      for (int j = 0; j < F_; ++j) dot += qlw[m * F_ + j] * fr[j];
      float x = tanhf((c[m] + dot) * inv) * 10.0f;
      if (!mask[(size_t)(b * M_ + m) * N_ + nn]) x = -__builtin_inff();
      ws_logits[(size_t)(b * M_ + m) * N_ + nn] = x;
    }
  }
}

// ---------------------------------------------------------------------------
// K5: per-batch softmax over M*N = 5000 -> probs
__global__ void __launch_bounds__(256)
k5_softmax(const float* __restrict__ ws_logits, float* __restrict__ out) {
  int b = blockIdx.x;
  __shared__ float red[8];
  const float* lg = ws_logits + (size_t)b * M_ * N_;
  float* ob = out + (size_t)b * M_ * N_;
  int tid = threadIdx.x;
  float lmax = -__builtin_inff();
  for (int i = tid; i < M_ * N_; i += 256) lmax = fmaxf(lmax, lg[i]);
  float gmax = block_max<8>(lmax, red);
  float ls = 0.f;
  for (int i = tid; i < M_ * N_; i += 256) ls += __expf(lg[i] - gmax);
  float S = block_sum<8>(ls, red);
  float invS = 1.f / S;
  for (int i = tid; i < M_ * N_; i += 256) ob[i] = __expf(lg[i] - gmax) * invS;
}

// ---------------------------------------------------------------------------
extern "C" void kernel_launch(void* const* d_in, const int* in_sizes, int n_in,
                              void* d_out, int out_size, void* d_ws, size_t ws_size,
                              hipStream_t stream) {
  // Input order per setup_inputs(); node_embeddings (d_in[0]) is unused.
  const float* fixed_context = (const float*)d_in[1];
  const float* prev          = (const float*)d_in[2];
  const float* ndf           = (const float*)d_in[3];
  const float* veh           = (const float*)d_in[4];
  const float* gV            = (const float*)d_in[5];
  const float* gK            = (const float*)d_in[6];
  const float* lK            = (const float*)d_in[7];
  const unsigned char* mask  = (const unsigned char*)d_in[8];  // numpy bool (1B)
  const float* W_pcv         = (const float*)d_in[9];
  const float* W_pns         = (const float*)d_in[10];
  const float* po            = (const float*)d_in[11];
  float* out = (float*)d_out;
  float* ws  = (float*)d_ws;

  // Workspace layout (floats), total ~12.1 MB
  float* ws_query  = ws;                       // B*M*D      = 40960
  float* ws_qkw    = ws_query + 40960;         // B*M*H*8    = 20480
  float* ws_compat = ws_qkw + 20480;           // H*B*M*N    = 2,560,000
  float* ws_concat = ws_compat + 2560000;      // B*M*D      = 40960
  float* ws_fq     = ws_concat + 40960;        // B*M*D      = 40960
  float* ws_qlw    = ws_fq + 40960;            // B*M*8      = 2560
  float* ws_logits = ws_qlw + 2560;            // B*M*N      = 320,000

  k1_query_qkw<<<B_ * M_, 128, 0, stream>>>(fixed_context, prev, veh, W_pcv,
                                            W_pns, ws_query, ws_qkw);
  k2_attention<<<H_ * B_, 128, 0, stream>>>(gK, gV, ndf, mask, W_pns, ws_query,
                                            ws_qkw, ws_compat, ws_concat);
  k3_finalq_qlw<<<B_ * M_, 128, 0, stream>>>(ws_concat, po, W_pns, ws_fq, ws_qlw);
  k4_logits<<<dim3(16, B_), 128, 0, stream>>>(lK, ndf, mask, ws_fq, ws_qlw,
                                              ws_logits);
  k5_softmax<<<B_, 256, 0, stream>>>(ws_logits, out);
}